// Sinkhorn_v3_21560735825972
// MI455X (gfx1250) — compile-verified
//
#include <hip/hip_runtime.h>

#define N 8192
#define SINK_EPS 1e-6f
#define ITERS 10

typedef __attribute__((ext_vector_type(2))) float v2f;
typedef __attribute__((ext_vector_type(8))) float v8f;
typedef __attribute__((ext_vector_type(4))) int v4i;
// Pointer-to-v4i in specific address spaces (HIP prints AS(1) as __device__).
typedef __attribute__((address_space(1))) v4i* gptr_v4i;
typedef __attribute__((address_space(3))) v4i* lptr_v4i;

__device__ __forceinline__ float wave_reduce_sum(float v) {
#pragma unroll
  for (int off = 16; off >= 1; off >>= 1)
    v += __shfl_xor(v, off, 32);
  return v;
}

// ---------------------------------------------------------------------------
// Kernel 0: initial row sums -> rinv[r] = 1 / max(sum|M[r,:]|, eps)
// wave-per-row streaming: lane strides float4 across the row (coalesced 512B/wave)
// grid = N/32 blocks of 256 threads; each wave handles 4 rows.
// ---------------------------------------------------------------------------
__global__ __launch_bounds__(256) void k_rowsum_inv(const float* __restrict__ M,
                                                    float* __restrict__ rinv) {
  const int lane = threadIdx.x & 31;
  const int wave = threadIdx.x >> 5;
  const int rowBase = blockIdx.x * 32 + wave * 4;
#pragma unroll
  for (int i = 0; i < 4; ++i) {
    const int r = rowBase + i;
    const float4* row = (const float4*)(M + (size_t)r * N);
    float4 acc = {0.f, 0.f, 0.f, 0.f};
#pragma unroll 4
    for (int ch = 0; ch < 64; ++ch) {
      float4 v = row[ch * 32 + lane];
      acc.x += fabsf(v.x); acc.y += fabsf(v.y);
      acc.z += fabsf(v.z); acc.w += fabsf(v.w);
    }
    float s = wave_reduce_sum(acc.x + acc.y + acc.z + acc.w);
    if (lane == 0) rinv[r] = 1.0f / fmaxf(s, SINK_EPS);
  }
}

// ---------------------------------------------------------------------------
// Pass A: dst = src * rinv[row]; accumulate column partial sums.
// Tile: 1024 cols (256 thr x float4) x 128 rows. Each thread owns 4 columns
// exclusively within its (bx,by) tile -> register accumulation, no atomics.
// grid = dim3(8, 64). partial[by][c] written once per tile (deterministic).
// ---------------------------------------------------------------------------
__global__ __launch_bounds__(256) void k_rowscale_colpart(const float* __restrict__ src,
                                                          float* __restrict__ dst,
                                                          const float* __restrict__ rinv,
                                                          float* __restrict__ partial) {
  const int c0 = blockIdx.x * 1024 + threadIdx.x * 4;
  const int r0 = blockIdx.y * 128;
  float4 acc = {0.f, 0.f, 0.f, 0.f};
  for (int i = 0; i < 128; ++i) {
    const int r = r0 + i;
    const float ri = rinv[r];  // uniform across block -> scalar load
    const size_t idx = (size_t)r * N + c0;
    float4 v = *(const float4*)(src + idx);
    v.x *= ri; v.y *= ri; v.z *= ri; v.w *= ri;
    *(float4*)(dst + idx) = v;
    acc.x += fabsf(v.x); acc.y += fabsf(v.y);
    acc.z += fabsf(v.z); acc.w += fabsf(v.w);
  }
  *(float4*)(partial + (size_t)blockIdx.y * N + c0) = acc;
}

// ---------------------------------------------------------------------------
// Column-partial reduction via WMMA: D = ones(16x4) x P(4x16) + C.
// Chaining C over 16 steps sums all 64 partial rows; with A == all-ones the
// result is invariant to the K-slot<->lane mapping, so this is layout-robust.
// One wave reduces 16 columns; grid = 8192/(16*8) = 64 blocks of 256.
// EXEC is all-ones throughout the WMMA region (no divergence before it).
// ---------------------------------------------------------------------------
__global__ __launch_bounds__(256) void k_colreduce_wmma(const float* __restrict__ partial,
                                                        float* __restrict__ cinv) {
  const int lane = threadIdx.x & 31;
  const int wave = threadIdx.x >> 5;
  const int cbase = (blockIdx.x * 8 + wave) * 16;
  const int col = cbase + (lane & 15);
  const int koff = (lane < 16) ? 0 : 2;

  v2f a; a.x = 1.0f; a.y = 1.0f;   // A = ones(16x4)
  v8f c = {0.f, 0.f, 0.f, 0.f, 0.f, 0.f, 0.f, 0.f};
#pragma unroll
  for (int s = 0; s < 16; ++s) {
    v2f b;
    b.x = partial[(size_t)(4 * s + koff) * N + col];
    b.y = partial[(size_t)(4 * s + koff + 1) * N + col];
    c = __builtin_amdgcn_wmma_f32_16x16x4_f32(
        /*neg_a=*/false, a, /*neg_b=*/false, b,
        /*c_mod=*/(short)0, c, /*reuse_a=*/false, /*reuse_b=*/false);
  }
  // Every row of D equals the column sums; VGPR0 lanes 0-15 hold (M=0, N=lane).
  if (lane < 16) cinv[col] = 1.0f / fmaxf(c[0], SINK_EPS);
}

// ---------------------------------------------------------------------------
// Pass B: M *= cinv[col] (in place), and compute next-iteration row sums.
// cinv (32KB) is broadcast into LDS once per block via the CDNA5 async
// global->LDS path, then each wave streams full rows (lane strides float4),
// reducing the row sum cross-lane once per row.
// grid = 256 blocks of 256 threads; 32 rows/block, 4 rows/wave.
// ---------------------------------------------------------------------------
__global__ __launch_bounds__(256) void k_colscale_rowsum(float* __restrict__ M,
                                                         const float* __restrict__ cinv,
                                                         float* __restrict__ rinv) {
  __shared__ __align__(16) float s_cinv[N];  // 32 KB of the 320 KB WGP LDS

#if __has_builtin(__builtin_amdgcn_global_load_async_to_lds_b128)
  {
    float* cinv_nc = const_cast<float*>(cinv);
#pragma unroll
    for (int i = 0; i < 8; ++i) {
      const int e = (i * 256 + threadIdx.x) * 4;  // float index, 16B granules
      __builtin_amdgcn_global_load_async_to_lds_b128(
          (gptr_v4i)(cinv_nc + e),
          (lptr_v4i)(s_cinv + e),
          0, 0);
    }
#if __has_builtin(__builtin_amdgcn_s_wait_asynccnt)
    __builtin_amdgcn_s_wait_asynccnt(0);
#else
    asm volatile("s_wait_asynccnt 0" ::: "memory");
#endif
  }
#else
  for (int i = threadIdx.x; i < N / 4; i += 256)
    ((float4*)s_cinv)[i] = ((const float4*)cinv)[i];
#endif
  __syncthreads();

  const int lane = threadIdx.x & 31;
  const int wave = threadIdx.x >> 5;
  const int rowBase = blockIdx.x * 32 + wave * 4;
#pragma unroll
  for (int i = 0; i < 4; ++i) {
    const int r = rowBase + i;
    float4* row = (float4*)(M + (size_t)r * N);
    const float4* sc = (const float4*)s_cinv;
    float4 acc = {0.f, 0.f, 0.f, 0.f};
#pragma unroll 4
    for (int ch = 0; ch < 64; ++ch) {
      const int e4 = ch * 32 + lane;
      float4 v = row[e4];
      float4 s = sc[e4];            // ds_load_b128 from LDS
      v.x *= s.x; v.y *= s.y; v.z *= s.z; v.w *= s.w;
      row[e4] = v;
      acc.x += fabsf(v.x); acc.y += fabsf(v.y);
      acc.z += fabsf(v.z); acc.w += fabsf(v.w);
    }
    float ssum = wave_reduce_sum(acc.x + acc.y + acc.z + acc.w);
    if (lane == 0) rinv[r] = 1.0f / fmaxf(ssum, SINK_EPS);
  }
}

// ---------------------------------------------------------------------------
// Host orchestration. d_in[0] = M (8192x8192 f32, never written).
// d_out = working + final buffer. d_ws: rinv(8192) | cinv(8192) | partial(64x8192)
// Total ws use ~2.1 MB. Fully deterministic (no atomics).
// ---------------------------------------------------------------------------
extern "C" void kernel_launch(void* const* d_in, const int* in_sizes, int n_in,
                              void* d_out, int out_size, void* d_ws, size_t ws_size,
                              hipStream_t stream) {
  (void)in_sizes; (void)n_in; (void)out_size; (void)ws_size;
  const float* Min = (const float*)d_in[0];
  float* M = (float*)d_out;
  float* rinv = (float*)d_ws;
  float* cinv = rinv + N;
  float* partial = cinv + N;

  k_rowsum_inv<<<N / 32, 256, 0, stream>>>(Min, rinv);

  const float* src = Min;
  for (int it = 0; it < ITERS; ++it) {
    k_rowscale_colpart<<<dim3(8, 64), 256, 0, stream>>>(src, M, rinv, partial);
    k_colreduce_wmma<<<64, 256, 0, stream>>>(partial, cinv);
    k_colscale_rowsum<<<256, 256, 0, stream>>>(M, cinv, rinv);
    src = M;  // subsequent iterations are in-place on d_out
  }
}